// HeteroRGCNLayer_50010599194657
// MI455X (gfx1250) — compile-verified
//
#include <hip/hip_runtime.h>
#include <cstdint>
#include <cstddef>

typedef __attribute__((ext_vector_type(2))) float v2f;
typedef __attribute__((ext_vector_type(4))) float v4f;
typedef __attribute__((ext_vector_type(8))) float v8f;

__device__ __forceinline__ float lrelu_f(float v) { return v >= 0.f ? v : 0.01f * v; }

// order-preserving float -> uint encoding for atomicMax
__device__ __forceinline__ unsigned enc_f(float f) {
  unsigned u = __float_as_uint(f);
  return (u & 0x80000000u) ? ~u : (u | 0x80000000u);
}
__device__ __forceinline__ float dec_f(unsigned k) {
  unsigned u = (k & 0x80000000u) ? (k & 0x7fffffffu) : ~k;
  return __uint_as_float(u);
}
#define ENC_NEG_INF 0x007FFFFFu  // enc_f(-inf)

// ---------------------------------------------------------------------------
// Generic fp32 WMMA GEMM:  out(MxN) = act( X(MxK) @ W(KxN) + bias + gadd[gidx[row]] )
// Block = 256 threads (8 wave32), 64 output rows per block.
// K in {256, 768}, N in {256, 768}. Uses V_WMMA_F32_16X16X4_F32.
// Loop order kk -> (a[4] from LDS) -> col-tile -> row-tile: each A fragment is
// loaded once per k-step and reused across all col tiles (CTW x fewer DS reads).
// ---------------------------------------------------------------------------
template <int K, int N>
__global__ __launch_bounds__(256) void gemm_wmma(
    const float* __restrict__ X, int M,
    const float* __restrict__ W,
    const float* __restrict__ bias,
    const float* __restrict__ gadd, const int* __restrict__ gidx,
    float* __restrict__ out, int act) {
  constexpr int KC  = 256;        // K chunk staged in LDS
  constexpr int NKC = K / KC;     // 1 or 3
  constexpr int CTW = N / 16 / 8; // col tiles per wave: 2 or 6
  constexpr int LDSROW = KC + 4;  // stride 260: 16B-aligned rows, conflict-free b64 reads

  __shared__ float Xs[64 * LDSROW];

  const int tid  = threadIdx.x;
  const int lane = tid & 31;
  const int wave = tid >> 5;
  const int row0 = blockIdx.x * 64;
  const int m    = lane & 15;          // M (or N) position within tile
  const int kp   = (lane >> 4) << 1;   // K pair base: 0 or 2

  v8f acc[CTW][4] = {};

  for (int kc = 0; kc < NKC; ++kc) {
    __syncthreads();
    // cooperative coalesced load of 64 x 256 chunk of X, 16B vectors
    for (int idx = tid; idx < 64 * (KC / 4); idx += 256) {
      int r  = idx >> 6;            // KC/4 == 64
      int c4 = (idx & 63) << 2;
      int gr = row0 + r;
      v4f v = {};
      if (gr < M) v = *(const v4f*)&X[(size_t)gr * K + kc * KC + c4];
      *(v4f*)&Xs[r * LDSROW + c4] = v;
    }
    __syncthreads();

    for (int kk = 0; kk < KC; kk += 4) {
      // A fragments (16x4): VGPR v holds K = kp + v, M = lane&15
      v2f a[4];
#pragma unroll
      for (int rt = 0; rt < 4; ++rt)
        a[rt] = *(const v2f*)&Xs[(rt * 16 + m) * LDSROW + kk + kp];
#pragma unroll
      for (int j = 0; j < CTW; ++j) {
        const int col = ((wave + j * 8) << 4) + m;
        // B fragment (4x16): VGPR v holds K = kp + v, N = lane&15
        const float* wp = &W[(size_t)(kc * KC + kk + kp) * N + col];
        v2f b;
        b.x = wp[0];
        b.y = wp[N];
#pragma unroll
        for (int rt = 0; rt < 4; ++rt) {
          acc[j][rt] = __builtin_amdgcn_wmma_f32_16x16x4_f32(
              false, a[rt], false, b, (short)0, acc[j][rt], false, false);
        }
      }
    }
  }

  // epilogue: C/D layout -> VGPR r: M = r + 8*(lane>=16), N = lane&15
#pragma unroll
  for (int j = 0; j < CTW; ++j) {
    const int col = ((wave + j * 8) << 4) + m;
    const float bc = bias ? bias[col] : 0.f;
#pragma unroll
    for (int rt = 0; rt < 4; ++rt) {
#pragma unroll
      for (int r8 = 0; r8 < 8; ++r8) {
        int row = row0 + rt * 16 + r8 + ((lane >> 4) << 3);
        if (row < M) {
          float v = acc[j][rt][r8] + bc;
          if (gadd) v += gadd[(size_t)gidx[row] * N + col];
          if (act) v = lrelu_f(v);
          out[(size_t)row * N + col] = v;
        }
      }
    }
  }
}

// ---------------------------------------------------------------------------
// span mean over bert rows (spans are <= 8 rows; direct sum beats cumsum)
// one block per edge, 256 threads over 768 cols
// ---------------------------------------------------------------------------
__global__ void span_mean_kernel(const float* __restrict__ bert,
                                 const int* __restrict__ span, int E,
                                 float* __restrict__ relE) {
  int e = blockIdx.x;
  if (e >= E) return;
  int s = span[2 * e], t = span[2 * e + 1];
  float inv = 1.0f / (float)(t - s);
  for (int c = threadIdx.x; c < 768; c += 256) {
    float accv = 0.f;
    for (int r = s; r < t; ++r) accv += bert[(size_t)r * 768 + c];
    relE[(size_t)e * 768 + c] = accv * inv;
  }
}

__global__ void fill_u32_kernel(unsigned* __restrict__ p, size_t n, unsigned v) {
  size_t i = (size_t)blockIdx.x * 256 + threadIdx.x;
  if (i < n) p[i] = v;
}

// ---------------------------------------------------------------------------
// attention passes: one block per edge, 256 threads = 256 channels
// ---------------------------------------------------------------------------
__global__ void att_max_reg(const int* __restrict__ src, const int* __restrict__ dst, int E,
                            const float* __restrict__ asrc, const float* __restrict__ adst,
                            const float* __restrict__ batt,
                            unsigned* __restrict__ mx, float* __restrict__ deg) {
  int e = blockIdx.x;
  if (e >= E) return;
  int s = src[e], d = dst[e], c = threadIdx.x;
  float v = lrelu_f(asrc[(size_t)s * 256 + c] + adst[(size_t)d * 256 + c] + batt[c]);
  atomicMax(&mx[(size_t)d * 256 + c], enc_f(v));
  if (c == 0) atomicAdd(&deg[d], 1.0f);
}

__global__ void att_max_rel(const int* __restrict__ dst, int E,
                            const float* __restrict__ ebuf,
                            unsigned* __restrict__ mx, float* __restrict__ deg) {
  int e = blockIdx.x;
  if (e >= E) return;
  int d = dst[e], c = threadIdx.x;
  float v = ebuf[(size_t)e * 256 + c];
  atomicMax(&mx[(size_t)d * 256 + c], enc_f(v));
  if (c == 0) atomicAdd(&deg[d], 1.0f);
}

__global__ void att_sum_reg(const int* __restrict__ src, const int* __restrict__ dst, int E,
                            const float* __restrict__ asrc, const float* __restrict__ adst,
                            const float* __restrict__ batt,
                            const float* __restrict__ msrc,
                            const unsigned* __restrict__ mx,
                            float* __restrict__ den, float* __restrict__ accb) {
  int e = blockIdx.x;
  if (e >= E) return;
  int s = src[e], d = dst[e], c = threadIdx.x;
  float v = lrelu_f(asrc[(size_t)s * 256 + c] + adst[(size_t)d * 256 + c] + batt[c]);
  float ex = __expf(v - dec_f(mx[(size_t)d * 256 + c]));
  atomicAdd(&den[(size_t)d * 256 + c], ex);
  atomicAdd(&accb[(size_t)d * 256 + c], ex * msrc[(size_t)s * 256 + c]);
}

__global__ void att_sum_rel(const int* __restrict__ dst, int E,
                            const float* __restrict__ ebuf, const float* __restrict__ mbuf,
                            const unsigned* __restrict__ mx,
                            float* __restrict__ den, float* __restrict__ accb) {
  int e = blockIdx.x;
  if (e >= E) return;
  int d = dst[e], c = threadIdx.x;
  float v = ebuf[(size_t)e * 256 + c];
  float ex = __expf(v - dec_f(mx[(size_t)d * 256 + c]));
  atomicAdd(&den[(size_t)d * 256 + c], ex);
  atomicAdd(&accb[(size_t)d * 256 + c], ex * mbuf[(size_t)e * 256 + c]);
}

__global__ void att_final(const float* __restrict__ acc1, const float* __restrict__ den1,
                          const float* __restrict__ acc2, const float* __restrict__ den2,
                          const float* __restrict__ deg, const float* __restrict__ fallback,
                          float* __restrict__ outp, int n) {
  int node = blockIdx.x;
  if (node >= n) return;
  size_t i = (size_t)node * 256 + threadIdx.x;
  float r = acc1[i] / fmaxf(den1[i], 1e-30f) + acc2[i] / fmaxf(den2[i], 1e-30f);
  outp[i] = (deg[node] > 0.f) ? r : fallback[i];
}

__global__ void scatter_add_kernel(const int* __restrict__ src, const int* __restrict__ dst,
                                   int E, const float* __restrict__ x, float* __restrict__ outp) {
  int e = blockIdx.x;
  if (e >= E) return;
  int c = threadIdx.x;
  atomicAdd(&outp[(size_t)dst[e] * 256 + c], x[(size_t)src[e] * 256 + c]);
}

// ---------------------------------------------------------------------------
// GRU pointwise: one block per row, 256 threads = 256 channels
// gh == nullptr -> use bhh broadcast (h==0 step); hin == nullptr -> h = 0
// ---------------------------------------------------------------------------
__global__ void gru_pointwise(const float* __restrict__ gi, const float* __restrict__ gh,
                              const float* __restrict__ bhh, const float* __restrict__ hin,
                              float* __restrict__ hout, int M) {
  int row = blockIdx.x;
  if (row >= M) return;
  int c = threadIdx.x;
  size_t b = (size_t)row * 768;
  float gir = gi[b + c], giz = gi[b + 256 + c], gin = gi[b + 512 + c];
  float ghr, ghz, ghn;
  if (gh) { ghr = gh[b + c]; ghz = gh[b + 256 + c]; ghn = gh[b + 512 + c]; }
  else    { ghr = bhh[c];    ghz = bhh[256 + c];    ghn = bhh[512 + c]; }
  float hp = hin ? hin[(size_t)row * 256 + c] : 0.f;
  float r = 1.f / (1.f + __expf(-(gir + ghr)));
  float z = 1.f / (1.f + __expf(-(giz + ghz)));
  float n = tanhf(gin + r * ghn);
  hout[(size_t)row * 256 + c] = (1.f - z) * n + z * hp;
}

// ---------------------------------------------------------------------------
extern "C" void kernel_launch(void* const* d_in, const int* in_sizes, int n_in,
                              void* d_out, int out_size, void* d_ws, size_t ws_size,
                              hipStream_t stream) {
  const float* feat_tok = (const float*)d_in[0];
  const float* feat_srl = (const float*)d_in[1];
  const float* feat_ent = (const float*)d_in[2];
  const float* bert     = (const float*)d_in[3];
  const float* W_rel    = (const float*)d_in[4];
  const float* b_rel    = (const float*)d_in[5];
  const float* W_nt1    = (const float*)d_in[6];
  const float* b_nt1    = (const float*)d_in[7];
  const float* W_nt2    = (const float*)d_in[8];
  const float* b_nt2    = (const float*)d_in[9];
  const float* W_att    = (const float*)d_in[10];
  const float* b_att    = (const float*)d_in[11];
  const float* Wih      = (const float*)d_in[12];
  const float* Whh      = (const float*)d_in[13];
  const float* bih      = (const float*)d_in[14];
  const float* bhh      = (const float*)d_in[15];
  const int* t2s_s = (const int*)d_in[16]; const int* t2s_d = (const int*)d_in[17];
  const int* s2s_s = (const int*)d_in[18]; const int* s2s_d = (const int*)d_in[19];
  const int* s2s_sp = (const int*)d_in[20];
  const int* e2e_s = (const int*)d_in[21]; const int* e2e_d = (const int*)d_in[22];
  const int* e2e_sp = (const int*)d_in[23];
  const int* t2e_s = (const int*)d_in[24]; const int* t2e_d = (const int*)d_in[25];
  const int* s2t_s = (const int*)d_in[26]; const int* s2t_d = (const int*)d_in[27];
  const int* e2t_s = (const int*)d_in[28]; const int* e2t_d = (const int*)d_in[29];

  const int NTOK = in_sizes[0] / 256;
  const int NSRL = in_sizes[1] / 256;
  const int NENT = in_sizes[2] / 256;
  const int ET2S = in_sizes[16], ES2S = in_sizes[18], EE2E = in_sizes[21];
  const int ET2E = in_sizes[24], ES2T = in_sizes[26], EE2T = in_sizes[28];

  const float* Wrel_top = W_rel;                 // 256x256
  const float* Wrel_bot = W_rel + 256 * 256;     // 768x256
  const float* Wa_top   = W_att;                 // 256x256
  const float* Wa_bot   = W_att + 256 * 256;     // 256x256

  float* out = (float*)d_out;
  float* srl_h = out + (size_t)NTOK * 256;
  float* ent_h = out + (size_t)(NTOK + NSRL) * 256;

  // ---- bump allocator in d_ws ----
  char* wsp = (char*)d_ws;
  auto alloc = [&](size_t nfloats) -> float* {
    float* p = (float*)wsp;
    wsp += ((nfloats * sizeof(float) + 255) / 256) * 256;
    return p;
  };
  const int EMAX = (ES2S > EE2E) ? ES2S : EE2E;
  int HM = NTOK;
  if (ES2S > HM) HM = ES2S;
  if (EE2E > HM) HM = EE2E;

  float* t_tok   = alloc((size_t)NTOK * 256);
  float* t_srl   = alloc((size_t)NSRL * 256);
  float* t_ent   = alloc((size_t)NENT * 256);
  float* a_src_tok = alloc((size_t)NTOK * 256);
  float* a_dst_srl = alloc((size_t)NSRL * 256);
  float* a_dst_ent = alloc((size_t)NENT * 256);
  float* relW_srl  = alloc((size_t)NSRL * 256);
  float* relW_ent  = alloc((size_t)NENT * 256);
  float* Htmp      = alloc((size_t)HM * 256);
  float* relE      = alloc((size_t)EMAX * 768);
  float* Y0        = alloc((size_t)EMAX * 256);
  float* m2 = alloc((size_t)ES2S * 256);
  float* e2 = alloc((size_t)ES2S * 256);
  float* m4 = alloc((size_t)EE2E * 256);
  float* e4 = alloc((size_t)EE2E * 256);
  unsigned* mx1s = (unsigned*)alloc((size_t)NSRL * 256);
  float* den1s = alloc((size_t)NSRL * 256);
  float* acc1s = alloc((size_t)NSRL * 256);
  unsigned* mx2s = (unsigned*)alloc((size_t)NSRL * 256);
  float* den2s = alloc((size_t)NSRL * 256);
  float* acc2s = alloc((size_t)NSRL * 256);
  unsigned* mx1e = (unsigned*)alloc((size_t)NENT * 256);
  float* den1e = alloc((size_t)NENT * 256);
  float* acc1e = alloc((size_t)NENT * 256);
  unsigned* mx2e = (unsigned*)alloc((size_t)NENT * 256);
  float* den2e = alloc((size_t)NENT * 256);
  float* acc2e = alloc((size_t)NENT * 256);
  float* deg_s = alloc((size_t)NSRL);
  float* deg_e = alloc((size_t)NENT);
  float* h_srl_tok = alloc((size_t)NTOK * 256);
  float* h_ent_tok = alloc((size_t)NTOK * 256);
  float* hbuf = alloc((size_t)NTOK * 256);
  const int CH = 25000;
  float* gi = alloc((size_t)CH * 768);
  float* gh = alloc((size_t)CH * 768);
  (void)ws_size; (void)n_in; (void)out_size;

  auto g256_256 = [&](const float* X, int M, const float* W, const float* bias,
                      const float* gadd, const int* gidx, float* o, int act) {
    gemm_wmma<256, 256><<<dim3((M + 63) / 64), dim3(256), 0, stream>>>(X, M, W, bias, gadd, gidx, o, act);
  };
  auto g768_256 = [&](const float* X, int M, const float* W, const float* bias,
                      const float* gadd, const int* gidx, float* o, int act) {
    gemm_wmma<768, 256><<<dim3((M + 63) / 64), dim3(256), 0, stream>>>(X, M, W, bias, gadd, gidx, o, act);
  };
  auto g256_768 = [&](const float* X, int M, const float* W, const float* bias, float* o) {
    gemm_wmma<256, 768><<<dim3((M + 63) / 64), dim3(256), 0, stream>>>(X, M, W, bias, nullptr, nullptr, o, 0);
  };
  auto node_trans = [&](const float* X, int M, float* o) {
    g256_256(X, M, W_nt1, b_nt1, nullptr, nullptr, Htmp, 1);
    g256_256(Htmp, M, W_nt2, b_nt2, nullptr, nullptr, o, 0);
  };

  // 1) per-node transforms (each node transformed once, not per edge)
  node_trans(feat_tok, NTOK, t_tok);
  node_trans(feat_srl, NSRL, t_srl);
  node_trans(feat_ent, NENT, t_ent);

  // 2) attention projections: concat@W_att = t@Wa_top (src side) + t@Wa_bot (dst side)
  g256_256(t_tok, NTOK, Wa_top, nullptr, nullptr, nullptr, a_src_tok, 0);
  g256_256(t_srl, NSRL, Wa_bot, nullptr, nullptr, nullptr, a_dst_srl, 0);
  g256_256(t_ent, NENT, Wa_bot, nullptr, nullptr, nullptr, a_dst_ent, 0);

  // 3) per-node half of W_rel
  g256_256(feat_srl, NSRL, Wrel_top, nullptr, nullptr, nullptr, relW_srl, 0);
  g256_256(feat_ent, NENT, Wrel_top, nullptr, nullptr, nullptr, relW_ent, 0);

  // 4) srl2srl relational messages
  span_mean_kernel<<<dim3(ES2S), dim3(256), 0, stream>>>(bert, s2s_sp, ES2S, relE);
  g768_256(relE, ES2S, Wrel_bot, b_rel, relW_srl, s2s_s, Y0, 0);
  node_trans(Y0, ES2S, m2);
  g256_256(m2, ES2S, Wa_top, b_att, a_dst_srl, s2s_d, e2, 1);

  // 5) ent2ent relational messages
  span_mean_kernel<<<dim3(EE2E), dim3(256), 0, stream>>>(bert, e2e_sp, EE2E, relE);
  g768_256(relE, EE2E, Wrel_bot, b_rel, relW_ent, e2e_s, Y0, 0);
  node_trans(Y0, EE2E, m4);
  g256_256(m4, EE2E, Wa_top, b_att, a_dst_ent, e2e_d, e4, 1);

  // 6) attention reduce into srl nodes (two separate softmaxes, summed)
  {
    size_t n = (size_t)NSRL * 256;
    fill_u32_kernel<<<dim3((unsigned)((n + 255) / 256)), dim3(256), 0, stream>>>(mx1s, n, ENC_NEG_INF);
    fill_u32_kernel<<<dim3((unsigned)((n + 255) / 256)), dim3(256), 0, stream>>>(mx2s, n, ENC_NEG_INF);
    hipMemsetAsync(den1s, 0, n * 4, stream); hipMemsetAsync(acc1s, 0, n * 4, stream);
    hipMemsetAsync(den2s, 0, n * 4, stream); hipMemsetAsync(acc2s, 0, n * 4, stream);
    hipMemsetAsync(deg_s, 0, (size_t)NSRL * 4, stream);
    att_max_reg<<<dim3(ET2S), dim3(256), 0, stream>>>(t2s_s, t2s_d, ET2S, a_src_tok, a_dst_srl, b_att, mx1s, deg_s);
    att_max_rel<<<dim3(ES2S), dim3(256), 0, stream>>>(s2s_d, ES2S, e2, mx2s, deg_s);
    att_sum_reg<<<dim3(ET2S), dim3(256), 0, stream>>>(t2s_s, t2s_d, ET2S, a_src_tok, a_dst_srl, b_att, t_tok, mx1s, den1s, acc1s);
    att_sum_rel<<<dim3(ES2S), dim3(256), 0, stream>>>(s2s_d, ES2S, e2, m2, mx2s, den2s, acc2s);
    att_final<<<dim3(NSRL), dim3(256), 0, stream>>>(acc1s, den1s, acc2s, den2s, deg_s, feat_srl, srl_h, NSRL);
  }

  // 7) attention reduce into ent nodes
  {
    size_t n = (size_t)NENT * 256;
    fill_u32_kernel<<<dim3((unsigned)((n + 255) / 256)), dim3(256), 0, stream>>>(mx1e, n, ENC_NEG_INF);
    fill_u32_kernel<<<dim3((unsigned)((n + 255) / 256)), dim3(256), 0, stream>>>(mx2e, n, ENC_NEG_INF);
    hipMemsetAsync(den1e, 0, n * 4, stream); hipMemsetAsync(acc1e, 0, n * 4, stream);
    hipMemsetAsync(den2e, 0, n * 4, stream); hipMemsetAsync(acc2e, 0, n * 4, stream);
    hipMemsetAsync(deg_e, 0, (size_t)NENT * 4, stream);
    att_max_reg<<<dim3(ET2E), dim3(256), 0, stream>>>(t2e_s, t2e_d, ET2E, a_src_tok, a_dst_ent, b_att, mx1e, deg_e);
    att_max_rel<<<dim3(EE2E), dim3(256), 0, stream>>>(e2e_d, EE2E, e4, mx2e, deg_e);
    att_sum_reg<<<dim3(ET2E), dim3(256), 0, stream>>>(t2e_s, t2e_d, ET2E, a_src_tok, a_dst_ent, b_att, t_tok, mx1e, den1e, acc1e);
    att_sum_rel<<<dim3(EE2E), dim3(256), 0, stream>>>(e2e_d, EE2E, e4, m4, mx2e, den2e, acc2e);
    att_final<<<dim3(NENT), dim3(256), 0, stream>>>(acc1e, den1e, acc2e, den2e, deg_e, feat_ent, ent_h, NENT);
  }

  // 8) scatter to tokens
  hipMemsetAsync(h_srl_tok, 0, (size_t)NTOK * 256 * 4, stream);
  hipMemsetAsync(h_ent_tok, 0, (size_t)NTOK * 256 * 4, stream);
  scatter_add_kernel<<<dim3(ES2T), dim3(256), 0, stream>>>(s2t_s, s2t_d, ES2T, srl_h, h_srl_tok);
  scatter_add_kernel<<<dim3(EE2T), dim3(256), 0, stream>>>(e2t_s, e2t_d, EE2T, ent_h, h_ent_tok);

  // 9) 3-step GRU over token rows, chunked to bound gi/gh workspace
  // step 1: h=0, x=h_srl_tok -> gh = b_hh broadcast (skip gh GEMM entirely)
  for (int r0 = 0; r0 < NTOK; r0 += CH) {
    int Mc = (NTOK - r0 < CH) ? (NTOK - r0) : CH;
    g256_768(h_srl_tok + (size_t)r0 * 256, Mc, Wih, bih, gi);
    gru_pointwise<<<dim3(Mc), dim3(256), 0, stream>>>(gi, nullptr, bhh, nullptr, hbuf + (size_t)r0 * 256, Mc);
  }
  // step 2: x=h_ent_tok
  for (int r0 = 0; r0 < NTOK; r0 += CH) {
    int Mc = (NTOK - r0 < CH) ? (NTOK - r0) : CH;
    g256_768(h_ent_tok + (size_t)r0 * 256, Mc, Wih, bih, gi);
    g256_768(hbuf + (size_t)r0 * 256, Mc, Whh, bhh, gh);
    gru_pointwise<<<dim3(Mc), dim3(256), 0, stream>>>(gi, gh, bhh, hbuf + (size_t)r0 * 256, hbuf + (size_t)r0 * 256, Mc);
  }
  // step 3: x = node_trans(feat_tok) = t_tok, write token output directly
  for (int r0 = 0; r0 < NTOK; r0 += CH) {
    int Mc = (NTOK - r0 < CH) ? (NTOK - r0) : CH;
    g256_768(t_tok + (size_t)r0 * 256, Mc, Wih, bih, gi);
    g256_768(hbuf + (size_t)r0 * 256, Mc, Whh, bhh, gh);
    gru_pointwise<<<dim3(Mc), dim3(256), 0, stream>>>(gi, gh, bhh, hbuf + (size_t)r0 * 256, out + (size_t)r0 * 256, Mc);
  }
}